// Quantizer_fp4_46265387713199
// MI455X (gfx1250) — compile-verified
//
#include <hip/hip_runtime.h>
#include <stdint.h>

// fp4 quantizer, MI455X (gfx1250).
//
// Reference: q = x/scale + zero; snap to codebook {0,2,3,4,4,5,6,8} (nearest,
// tie -> lower code); out = (code - zero)*scale.  Pure elementwise stream:
// 268 MB @ 23.3 TB/s => ~11.5 us floor; WMMA inapplicable (no contraction).
//
// Quantizer: interior codes {2..6} are consecutive ints with round-half-down:
//   code = clamp(ceil(q - 0.5), 2, 6); outer codes via row-uniform x-space
//   boundaries (scale > 0): x <= (1-z)s -> 0 ; x > (7-z)s -> 8.
// => 9 VALU ops/element, under the ~2-3 lane-ops/byte machine balance.
//
// Front end: CDNA5 async global->LDS staging (ASYNCcnt), 4 buffers, 3 in
// flight per wave. Each lane consumes exactly the LDS bytes its own async
// load wrote -> the wave's own s_wait_asynccnt is the only sync needed.
// Both the async write and the read-back use the same raw-offset AS3
// provenance so alias analysis cannot fold the staging away.

typedef float f4 __attribute__((ext_vector_type(4)));
typedef int   i4 __attribute__((vector_size(16)));   // matches builtin param

#define QCOLS 8192
#define TPB   256                      // 8 wave32s per block
#define F4_PER_ROW (QCOLS / 4)         // 2048
#define NCHUNK (F4_PER_ROW / TPB)      // 8 chunks of 256 lanes x 16 B
#define DEPTH  4                       // LDS staging buffers (16 KB total)

#if defined(__HIP_DEVICE_COMPILE__) && \
    __has_builtin(__builtin_amdgcn_global_load_async_to_lds_b128)
#define USE_ASYNC 1
#else
#define USE_ASYNC 0
#endif

struct QP { float inv_s, zm, s, m, b_lo, b_hi, c_hi; };

__device__ __forceinline__ float quant1(float v, const QP& p) {
  float t = __builtin_fmaf(v, p.inv_s, p.zm);   // q - 0.5 (no per-elem divide)
  float r = __builtin_ceilf(t);                 // round-half-down code
  r = fminf(fmaxf(r, 2.0f), 6.0f);              // interior codes {2..6}
  float o = __builtin_fmaf(r, p.s, p.m);        // dequantize
  o = (v > p.b_hi) ? p.c_hi : o;                // code 8 region
  o = (v <= p.b_lo) ? p.m : o;                  // code 0 region
  return o;
}

__device__ __forceinline__ f4 quant4(f4 v, const QP& p) {
  f4 r;
  r.x = quant1(v.x, p); r.y = quant1(v.y, p);
  r.z = quant1(v.z, p); r.w = quant1(v.w, p);
  return r;
}

#if USE_ASYNC
typedef __attribute__((address_space(1))) i4 as1_i4;  // global (__device__)
typedef __attribute__((address_space(3))) i4 as3_i4;  // LDS (__shared__)
typedef __attribute__((address_space(3))) f4 as3_f4;  // LDS, float view

__device__ __forceinline__ uint32_t lds_off(int chunk) {
  return (uint32_t)((((chunk % DEPTH) * TPB) + threadIdx.x) * 16u);
}

template <int C>
__device__ __forceinline__ void issue_chunk(const f4* xv) {
  // AS1 ptr: global flat address is numerically the AS1 address.
  as1_i4* gp = (as1_i4*)(uintptr_t)(xv + C * TPB + threadIdx.x);
  // AS3 ptr: raw LDS byte offset (single __shared__ array at base offset 0).
  as3_i4* lp = (as3_i4*)lds_off(C);
  __builtin_amdgcn_global_load_async_to_lds_b128(gp, lp, 0, 0);
}

template <int N>
__device__ __forceinline__ void wait_async() {
  // "memory" clobber: the LDS read-back may not be hoisted above this wait.
  asm volatile("s_wait_asynccnt %0" ::"n"(N) : "memory");
}

template <int C>
__device__ __forceinline__ void step(const f4* xv, f4* ov, const QP& p) {
  if constexpr (C + DEPTH - 1 < NCHUNK) issue_chunk<C + DEPTH - 1>(xv);
  // After issuing, chunks <= C are done once <= N remain outstanding.
  wait_async<(C <= NCHUNK - DEPTH) ? (DEPTH - 1) : (NCHUNK - 1 - C)>();
  // Read back through the SAME raw-offset AS3 provenance as the async write
  // (may-alias => not foldable) -> ds_load_b128.
  f4 v = *(const as3_f4*)lds_off(C);
  __builtin_nontemporal_store(quant4(v, p), ov + C * TPB + threadIdx.x);
}
#endif

__global__ __launch_bounds__(TPB) void fp4_quant_rows(
    const float* __restrict__ x, const float* __restrict__ scale,
    const float* __restrict__ zero, float* __restrict__ out) {
  const uint32_t row = blockIdx.x;

  // Uniform per-row scalars -> s_load + SALU-float precompute.
  const float s = scale[row];
  const float z = zero[row];
  const QP p = {1.0f / s,        z - 0.5f,        s,
                -z * s,          (1.0f - z) * s,  (7.0f - z) * s,
                (8.0f - z) * s};

  const size_t base = (size_t)row * QCOLS;
  const f4* __restrict__ xv = (const f4*)(x + base);
  f4* __restrict__ ov = (f4*)(out + base);

  // Warm L2 for the row's tail while the pipeline spins up.
  __builtin_prefetch(xv + (NCHUNK / 2) * TPB + threadIdx.x, 0, 0);

#if USE_ASYNC
  // 16 KB LDS reservation for the staging buffers. Escape its address so it
  // stays allocated and is covered by the waits' memory clobbers.
  __shared__ f4 stage[DEPTH * TPB];
  asm volatile("" ::"v"((uint32_t)(uintptr_t)stage) : "memory");

  issue_chunk<0>(xv);
  issue_chunk<1>(xv);
  issue_chunk<2>(xv);
  step<0>(xv, ov, p);
  step<1>(xv, ov, p);
  step<2>(xv, ov, p);
  step<3>(xv, ov, p);
  step<4>(xv, ov, p);
  step<5>(xv, ov, p);
  step<6>(xv, ov, p);
  step<7>(xv, ov, p);
#else
#pragma unroll
  for (int it = 0; it < NCHUNK; ++it) {
    const uint32_t i = it * TPB + threadIdx.x;
    f4 v = __builtin_nontemporal_load(xv + i);
    __builtin_nontemporal_store(quant4(v, p), ov + i);
  }
#endif
}

extern "C" void kernel_launch(void* const* d_in, const int* in_sizes, int n_in,
                              void* d_out, int out_size, void* d_ws,
                              size_t ws_size, hipStream_t stream) {
  (void)n_in; (void)out_size; (void)d_ws; (void)ws_size;
  const float* x     = (const float*)d_in[0];
  const float* scale = (const float*)d_in[1];
  const float* zero  = (const float*)d_in[2];
  float* out         = (float*)d_out;

  const int rows = in_sizes[1];  // one scale per row
  fp4_quant_rows<<<rows, TPB, 0, stream>>>(x, scale, zero, out);
}